// TestNet_24068996727264
// MI455X (gfx1250) — compile-verified
//
#include <hip/hip_runtime.h>
#include <math.h>

// ---------------------------------------------------------------------------
// TestNet: 2x graph-conv (edge scatter-add) -> fc1 -> fc2 GEMM -> softmax
// fc2 (256x58500 @ 58500x64) implemented with V_WMMA_F32_16X16X4_F32 (exact fp32).
// ---------------------------------------------------------------------------

typedef __attribute__((ext_vector_type(2))) float v2f;
typedef __attribute__((ext_vector_type(8))) float v8f;

#define B_MESH   256
#define VPM      5850
#define NV       (B_MESH * VPM)     // 1,497,600 vertices
#define NE       (3 * NV)           // 4,492,800 edges
#define K_FC2    (VPM * 10)         // 58,500
#define NEG_SLOPE 0.01f
#define KCHUNKS  25
#define KCHUNK   (K_FC2 / KCHUNKS)  // 2,340 (multiple of 4)

__device__ __forceinline__ float leaky(float t) {
  return t >= 0.0f ? t : NEG_SLOPE * t;
}

// ---- graph conv layer 1: verts(V,3) -> out(V,5), nbr(V,5), biases included
__global__ void gc1_kernel(const float* __restrict__ verts,
                           const float* __restrict__ w0, const float* __restrict__ b0,
                           const float* __restrict__ w1, const float* __restrict__ b1,
                           float* __restrict__ out, float* __restrict__ nbr) {
  int v = blockIdx.x * blockDim.x + threadIdx.x;
  if (v >= NV) return;
  float x0 = verts[3 * (size_t)v + 0];
  float x1 = verts[3 * (size_t)v + 1];
  float x2 = verts[3 * (size_t)v + 2];
#pragma unroll
  for (int c = 0; c < 5; ++c) {
    out[(size_t)v * 5 + c] = b0[c] + x0 * w0[c * 3 + 0] + x1 * w0[c * 3 + 1] + x2 * w0[c * 3 + 2];
    nbr[(size_t)v * 5 + c] = b1[c] + x0 * w1[c * 3 + 0] + x1 * w1[c * 3 + 1] + x2 * w1[c * 3 + 2];
  }
}

// ---- edge scatter-add, 5 channels: out[i] += nbr[j]; out[j] += nbr[i]
__global__ void scatter1_kernel(const int* __restrict__ edges,
                                const float* __restrict__ nbr,
                                float* __restrict__ out) {
  int e = blockIdx.x * blockDim.x + threadIdx.x;
  if (e >= NE) return;
  int i = edges[2 * (size_t)e + 0];
  int j = edges[2 * (size_t)e + 1];
#pragma unroll
  for (int c = 0; c < 5; ++c) {
    atomicAdd(&out[(size_t)i * 5 + c], nbr[(size_t)j * 5 + c]);
    atomicAdd(&out[(size_t)j * 5 + c], nbr[(size_t)i * 5 + c]);
  }
}

// ---- graph conv layer 2: x = leaky(out1); -> out(V,20), nbr(V,20)
__global__ void gc2_kernel(const float* __restrict__ x1raw,
                           const float* __restrict__ w0, const float* __restrict__ b0,
                           const float* __restrict__ w1, const float* __restrict__ b1,
                           float* __restrict__ out, float* __restrict__ nbr) {
  int v = blockIdx.x * blockDim.x + threadIdx.x;
  if (v >= NV) return;
  float x[5];
#pragma unroll
  for (int d = 0; d < 5; ++d) x[d] = leaky(x1raw[(size_t)v * 5 + d]);
#pragma unroll
  for (int c = 0; c < 20; ++c) {
    float s0 = b0[c], s1 = b1[c];
#pragma unroll
    for (int d = 0; d < 5; ++d) {
      s0 += x[d] * w0[c * 5 + d];
      s1 += x[d] * w1[c * 5 + d];
    }
    out[(size_t)v * 20 + c] = s0;
    nbr[(size_t)v * 20 + c] = s1;
  }
}

// ---- edge scatter-add, 20 channels
__global__ void scatter2_kernel(const int* __restrict__ edges,
                                const float* __restrict__ nbr,
                                float* __restrict__ out) {
  int e = blockIdx.x * blockDim.x + threadIdx.x;
  if (e >= NE) return;
  int i = edges[2 * (size_t)e + 0];
  int j = edges[2 * (size_t)e + 1];
#pragma unroll
  for (int c = 0; c < 20; ++c) {
    atomicAdd(&out[(size_t)i * 20 + c], nbr[(size_t)j * 20 + c]);
    atomicAdd(&out[(size_t)j * 20 + c], nbr[(size_t)i * 20 + c]);
  }
}

// ---- fc1: x = leaky(out2); y = leaky(x @ fc1_w.T + fc1_b) -> Y(V,10)
__global__ void fc1_kernel(const float* __restrict__ x2raw,
                           const float* __restrict__ fw, const float* __restrict__ fb,
                           float* __restrict__ Y) {
  int v = blockIdx.x * blockDim.x + threadIdx.x;
  if (v >= NV) return;
  float x[20];
#pragma unroll
  for (int d = 0; d < 20; ++d) x[d] = leaky(x2raw[(size_t)v * 20 + d]);
#pragma unroll
  for (int c = 0; c < 10; ++c) {
    float s = fb[c];
#pragma unroll
    for (int d = 0; d < 20; ++d) s += x[d] * fw[c * 20 + d];
    Y[(size_t)v * 10 + c] = leaky(s);
  }
}

// ---- fc2 GEMM tile via V_WMMA_F32_16X16X4_F32.
// Y viewed as A = (256 x 58500) row-major; B = fc2_w.T, i.e. B[k][n] = W[n*K + k].
// One wave per (colTile, rowTile, kChunk). Deterministic partial outputs.
// A/B lane layout for 16x16x4 f32 (ISA 7.12.2): lane L holds
//   A[M = L&15][k + 2*(L>>4) + {0,1}]  and  B[k + 2*(L>>4) + {0,1}][N = L&15].
__global__ void fc2_wmma_kernel(const float* __restrict__ Y,
                                const float* __restrict__ W,
                                float* __restrict__ partial) {
  const int lane    = threadIdx.x;   // 0..31 (wave32)
  const int colTile = blockIdx.x;    // 0..3
  const int rowTile = blockIdx.y;    // 0..15
  const int chunk   = blockIdx.z;    // 0..KCHUNKS-1
  const int mn   = lane & 15;
  const int kOff = (lane >> 4) << 1; // 0 or 2

  const float* arow = Y + (size_t)(rowTile * 16 + mn) * K_FC2;
  const float* brow = W + (size_t)(colTile * 16 + mn) * K_FC2;

  v8f c = {};
  const int kBase = chunk * KCHUNK;
  for (int k = kBase; k < kBase + KCHUNK; k += 4) {
    v2f a = *(const v2f*)(arow + k + kOff);
    v2f b = *(const v2f*)(brow + k + kOff);
    // (neg_a, A, neg_b, B, c_mod, C, reuse_a, reuse_b)
    c = __builtin_amdgcn_wmma_f32_16x16x4_f32(false, a, false, b, (short)0, c,
                                              false, false);
  }

  // C/D layout: lane L covers N = L&15, rows M = r + 8*(L>>4), r = VGPR index.
  float* pt = partial + (size_t)chunk * (B_MESH * 64);
  const int col = colTile * 16 + (lane & 15);
  const int rhi = (lane >> 4) << 3;
#pragma unroll
  for (int r = 0; r < 8; ++r) {
    int row = rowTile * 16 + r + rhi;
    pt[(size_t)row * 64 + col] = c[r];
  }
}

// ---- fixed-order K-chunk reduction + bias + softmax; one wave per mesh row
__global__ void softmax_kernel(const float* __restrict__ partial,
                               const float* __restrict__ bias,
                               float* __restrict__ out) {
  const int row  = blockIdx.x;   // 0..255
  const int lane = threadIdx.x;  // 0..31
  float v0 = bias[lane];
  float v1 = bias[lane + 32];
  for (int ch = 0; ch < KCHUNKS; ++ch) {
    const float* p = partial + (size_t)ch * (B_MESH * 64) + (size_t)row * 64;
    v0 += p[lane];
    v1 += p[lane + 32];
  }
  float m = fmaxf(v0, v1);
#pragma unroll
  for (int off = 16; off >= 1; off >>= 1) m = fmaxf(m, __shfl_xor(m, off, 32));
  float e0 = expf(v0 - m);
  float e1 = expf(v1 - m);
  float s = e0 + e1;
#pragma unroll
  for (int off = 16; off >= 1; off >>= 1) s += __shfl_xor(s, off, 32);
  float inv = 1.0f / s;
  out[(size_t)row * 64 + lane]      = e0 * inv;
  out[(size_t)row * 64 + lane + 32] = e1 * inv;
}

extern "C" void kernel_launch(void* const* d_in, const int* in_sizes, int n_in,
                              void* d_out, int out_size, void* d_ws, size_t ws_size,
                              hipStream_t stream) {
  const float* verts = (const float*)d_in[0];
  const int*   edges = (const int*)d_in[1];
  const float* w0a   = (const float*)d_in[2];
  const float* b0a   = (const float*)d_in[3];
  const float* w1a   = (const float*)d_in[4];
  const float* b1a   = (const float*)d_in[5];
  const float* w0b   = (const float*)d_in[6];
  const float* b0b   = (const float*)d_in[7];
  const float* w1b   = (const float*)d_in[8];
  const float* b1b   = (const float*)d_in[9];
  const float* fc1_w = (const float*)d_in[10];
  const float* fc1_b = (const float*)d_in[11];
  const float* fc2_w = (const float*)d_in[12];
  const float* fc2_b = (const float*)d_in[13];

  float* ws      = (float*)d_ws;
  float* out1    = ws;                       // NV*5
  float* nbr1    = ws + (size_t)NV * 5;      // NV*5
  float* out2    = ws + (size_t)NV * 10;     // NV*20
  float* nbr2    = ws + (size_t)NV * 30;     // NV*20
  float* Y       = ws;                       // NV*10 (reuses dead out1+nbr1)
  float* partial = ws + (size_t)NV * 50;     // KCHUNKS*256*64

  const int TPB = 256;
  gc1_kernel<<<(NV + TPB - 1) / TPB, TPB, 0, stream>>>(verts, w0a, b0a, w1a, b1a, out1, nbr1);
  scatter1_kernel<<<(NE + TPB - 1) / TPB, TPB, 0, stream>>>(edges, nbr1, out1);
  gc2_kernel<<<(NV + TPB - 1) / TPB, TPB, 0, stream>>>(out1, w0b, b0b, w1b, b1b, out2, nbr2);
  scatter2_kernel<<<(NE + TPB - 1) / TPB, TPB, 0, stream>>>(edges, nbr2, out2);
  fc1_kernel<<<(NV + TPB - 1) / TPB, TPB, 0, stream>>>(out2, fc1_w, fc1_b, Y);
  fc2_wmma_kernel<<<dim3(4, 16, KCHUNKS), 32, 0, stream>>>(Y, fc2_w, partial);
  softmax_kernel<<<B_MESH, 32, 0, stream>>>(partial, fc2_b, (float*)d_out);
}